// graph_constructor_13941463843416
// MI455X (gfx1250) — compile-verified
//
#include <hip/hip_runtime.h>
#include <hip/hip_bf16.h>
#include <math.h>

#define NNODES 6144
#define DIM    64
#define LAYERS 3
#define TOPK   24
#define ALPHA  3.0f

typedef __attribute__((ext_vector_type(2))) float v2f;
typedef __attribute__((ext_vector_type(8))) float v8f;

// ---------------------------------------------------------------------------
// 1) gather: V1 = emb1[idx], V2 = emb2[idx]
// ---------------------------------------------------------------------------
__global__ void gather_kernel(const int* __restrict__ idx,
                              const float* __restrict__ emb1,
                              const float* __restrict__ emb2,
                              float* __restrict__ v1,
                              float* __restrict__ v2, int n) {
  int t = blockIdx.x * blockDim.x + threadIdx.x;
  if (t >= n * DIM) return;
  int r = t / DIM, d = t % DIM;
  int src = idx[r];
  v1[t] = emb1[src * DIM + d];
  v2[t] = emb2[src * DIM + d];
}

// ---------------------------------------------------------------------------
// 2) per-layer MLP update: vn = tanh(ALPHA * (scale * vc @ W^T + b))
//    blockIdx.y selects which state (0 -> {v1,W1,b1}, 1 -> {v2,W2,b2})
// ---------------------------------------------------------------------------
__global__ void mlp_kernel(const float* __restrict__ v1c, const float* __restrict__ v2c,
                           float* __restrict__ v1n, float* __restrict__ v2n,
                           const float* __restrict__ W1, const float* __restrict__ b1,
                           const float* __restrict__ W2, const float* __restrict__ b2,
                           const float* __restrict__ scale_set, int layer, int n) {
  int t = blockIdx.x * blockDim.x + threadIdx.x;
  if (t >= n * DIM) return;
  int r = t / DIM, o = t % DIM;
  float s = scale_set[layer];
  const float* vc = (blockIdx.y == 0) ? v1c : v2c;
  const float* W  = ((blockIdx.y == 0) ? W1 : W2) + layer * DIM * DIM + o * DIM;
  const float* b  = ((blockIdx.y == 0) ? b1 : b2) + layer * DIM;
  float*       vn = (blockIdx.y == 0) ? v1n : v2n;
  const float* row = vc + r * DIM;
  float acc = 0.0f;
#pragma unroll 8
  for (int d = 0; d < DIM; ++d) acc += row[d] * W[d];
  vn[t] = tanhf(ALPHA * (s * acc + b[o]));
}

// ---------------------------------------------------------------------------
// 3) adjacency: out[i][j] = relu(tanh(ALPHA*(V1[i].V2[j] - V2[i].V1[j])))
//    64x64 macro tile per 256-thread block, WMMA f32 16x16x4, K = 64.
//    LDS pitch 66 floats -> conflict-free column-strided reads, float2 aligned.
// ---------------------------------------------------------------------------
#define PITCH 66

__global__ void adj_kernel(const float* __restrict__ V1,
                           const float* __restrict__ V2,
                           float* __restrict__ outL, int n) {
  extern __shared__ float smem[];            // 4 * 64 * PITCH floats
  float* sV1I = smem;
  float* sV2I = smem + 1 * 64 * PITCH;
  float* sV1J = smem + 2 * 64 * PITCH;
  float* sV2J = smem + 3 * 64 * PITCH;

  const int Ibase = blockIdx.y * 64;
  const int Jbase = blockIdx.x * 64;
  const int tid = threadIdx.x;

  // cooperative load of the four 64x64 tiles
  for (int e = tid; e < 64 * DIM; e += 256) {
    int row = e / DIM, col = e % DIM;
    sV1I[row * PITCH + col] = V1[(Ibase + row) * DIM + col];
    sV2I[row * PITCH + col] = V2[(Ibase + row) * DIM + col];
    sV1J[row * PITCH + col] = V1[(Jbase + row) * DIM + col];
    sV2J[row * PITCH + col] = V2[(Jbase + row) * DIM + col];
  }
  __syncthreads();

  const int wave = tid >> 5;
  const int lane = tid & 31;
  const int half = lane >> 4;      // K-group select for 16x4 f32 operand layout
  const int lr   = lane & 15;      // M (for A) / N (for B) within the operand

  for (int s = wave; s < 16; s += 8) {     // 2 of the 16 subtiles per wave
    const int Moff = (s >> 2) * 16;
    const int Noff = (s & 3) * 16;
    v8f acc1 = {0.f, 0.f, 0.f, 0.f, 0.f, 0.f, 0.f, 0.f};
    v8f acc2 = {0.f, 0.f, 0.f, 0.f, 0.f, 0.f, 0.f, 0.f};
#pragma unroll
    for (int k = 0; k < DIM; k += 4) {
      const int col = k + 2 * half;
      v2f a1 = *(const v2f*)&sV1I[(Moff + lr) * PITCH + col];
      v2f b1 = *(const v2f*)&sV2J[(Noff + lr) * PITCH + col];
      acc1 = __builtin_amdgcn_wmma_f32_16x16x4_f32(false, a1, false, b1,
                                                   (short)0, acc1, false, false);
      v2f a2 = *(const v2f*)&sV2I[(Moff + lr) * PITCH + col];
      v2f b2 = *(const v2f*)&sV1J[(Noff + lr) * PITCH + col];
      acc2 = __builtin_amdgcn_wmma_f32_16x16x4_f32(false, a2, false, b2,
                                                   (short)0, acc2, false, false);
    }
    // C/D layout: VGPR v, lane<16 -> row v, col=lane; lane>=16 -> row v+8
#pragma unroll
    for (int v = 0; v < 8; ++v) {
      const int m = v + 8 * half;
      float a = acc1[v] - acc2[v];
      float t = tanhf(ALPHA * a);
      float r = t > 0.0f ? t : 0.0f;
      outL[(size_t)(Ibase + Moff + m) * n + (Jbase + Noff + lr)] = r;
    }
  }
}

// ---------------------------------------------------------------------------
// 4) per-row top-K mask, in place. Row of 6144 -> 24 elems/thread (256 thr).
//    Values are in [0,1) (relu o tanh) so float order == uint bit order.
// ---------------------------------------------------------------------------
__global__ void topk_kernel(float* __restrict__ outL, int n) {
  const int row = blockIdx.x;
  const int tid = threadIdx.x;
  float* p = outL + (size_t)row * n;

  float    x[24];
  unsigned u[24];
#pragma unroll
  for (int j = 0; j < 24; ++j) {
    x[j] = p[j * 256 + tid];
    u[j] = __float_as_uint(x[j]);
  }

  __shared__ int red[256];
  unsigned lo = 0u, hi = 0x3F800000u;      // [0, 1.0)
  for (int it = 0; it < 30; ++it) {
    unsigned mid = (lo + hi) >> 1;
    int c = 0;
#pragma unroll
    for (int j = 0; j < 24; ++j) c += (u[j] > mid);
    red[tid] = c;
    __syncthreads();
    for (int ofs = 128; ofs > 0; ofs >>= 1) {
      if (tid < ofs) red[tid] += red[tid + ofs];
      __syncthreads();
    }
    int total = red[0];
    __syncthreads();
    if (total >= TOPK) lo = mid; else hi = mid;
  }
#pragma unroll
  for (int j = 0; j < 24; ++j)
    p[j * 256 + tid] = (u[j] > lo) ? x[j] : 0.0f;
}

// ---------------------------------------------------------------------------
extern "C" void kernel_launch(void* const* d_in, const int* in_sizes, int n_in,
                              void* d_out, int out_size, void* d_ws, size_t ws_size,
                              hipStream_t stream) {
  const int*   idx       = (const int*)  d_in[0];
  // d_in[1] = scale_idx (unused by the reference body)
  const float* scale_set = (const float*)d_in[2];
  const float* emb1      = (const float*)d_in[3];
  const float* emb2      = (const float*)d_in[4];
  const float* W1        = (const float*)d_in[5];
  const float* b1        = (const float*)d_in[6];
  const float* W2        = (const float*)d_in[7];
  const float* b2        = (const float*)d_in[8];
  float* out = (float*)d_out;

  const int n = NNODES;
  const size_t vlen = (size_t)n * DIM;

  // ping-pong node-state buffers in workspace (4 * 1.57 MB)
  float* v1a = (float*)d_ws;
  float* v2a = v1a + vlen;
  float* v1b = v2a + vlen;
  float* v2b = v1b + vlen;

  dim3 blk(256);
  dim3 gthr((n * DIM + 255) / 256);

  gather_kernel<<<gthr, blk, 0, stream>>>(idx, emb1, emb2, v1a, v2a, n);

  float* v1c = v1a; float* v2c = v2a;
  float* v1n = v1b; float* v2n = v2b;

  const size_t smem_bytes = 4u * 64u * PITCH * sizeof(float);  // 67584 B

  for (int l = 0; l < LAYERS; ++l) {
    mlp_kernel<<<dim3(gthr.x, 2), blk, 0, stream>>>(v1c, v2c, v1n, v2n,
                                                    W1, b1, W2, b2,
                                                    scale_set, l, n);
    // swap state
    float* t1 = v1c; v1c = v1n; v1n = t1;
    float* t2 = v2c; v2c = v2n; v2n = t2;

    float* outL = out + (size_t)l * n * n;
    adj_kernel<<<dim3(n / 64, n / 64), blk, smem_bytes, stream>>>(v1c, v2c, outL, n);
    topk_kernel<<<dim3(n), blk, 0, stream>>>(outL, n);
  }
}